// trainLoss_18769007084480
// MI455X (gfx1250) — compile-verified
//
#include <hip/hip_runtime.h>
#include <hip/hip_bf16.h>

#define Bn 32768
#define Dn 2048
#define TYPEN 5
#define CEPS 1e-8f

#define ROWS_PER_CHUNK 256
#define NCHUNK (Bn / ROWS_PER_CHUNK) /* 128 */

// ---- workspace layout (element offsets into a float/int view of d_ws) ----
#define WS_Y      0                                   // int   [Bn]
#define WS_CE     (Bn)                                // float [Bn]
#define WS_PART   (2 * Bn)                            // float [NCHUNK*TYPEN*Dn]
#define WS_CNT    (WS_PART + NCHUNK * TYPEN * Dn)     // int   [8]
#define WS_CENTER (WS_CNT + 16)                       // float [TYPEN*Dn]
#define WS_CT     (WS_CENTER + TYPEN * Dn + 16)       // float [(Bn/16)*256]
#define WS_NORMX  (WS_CT + (Bn / 16) * 256)           // float [Bn]

typedef __attribute__((ext_vector_type(2))) float v2f;
typedef __attribute__((ext_vector_type(4))) float v4f;
typedef __attribute__((ext_vector_type(8))) float v8f;

// ---------------------------------------------------------------- init ----
__global__ void init_kernel(int* cnt) {
    if (threadIdx.x < 8) cnt[threadIdx.x] = 0;
}

// ------------------------------------------------- per-row prep (CE, y) ---
__global__ void prep_kernel(const float* __restrict__ oh,
                            const float* __restrict__ th,
                            int* __restrict__ y,
                            float* __restrict__ ce,
                            int* __restrict__ counts) {
    const int row = blockIdx.x * blockDim.x + threadIdx.x;
    if (row >= Bn) return;
    float o[TYPEN], t[TYPEN];
#pragma unroll
    for (int c = 0; c < TYPEN; ++c) {
        o[c] = oh[row * TYPEN + c];
        t[c] = th[row * TYPEN + c];
    }
    float m = o[0];
#pragma unroll
    for (int c = 1; c < TYPEN; ++c) m = fmaxf(m, o[c]);
    float se = 0.f;
#pragma unroll
    for (int c = 0; c < TYPEN; ++c) se += expf(o[c] - m);
    const float lse = logf(se);
    float s = 0.f;
#pragma unroll
    for (int c = 0; c < TYPEN; ++c) s += t[c] * (o[c] - m - lse);
    ce[row] = -s;

    int best = 0;
    float bv = t[0];
#pragma unroll
    for (int c = 1; c < TYPEN; ++c) {
        if (t[c] > bv) { bv = t[c]; best = c; }   // first-max tie break, like jnp.argmax
    }
    y[row] = best;
    atomicAdd(&counts[best], 1);                  // integer atomic: order-independent
}

// --------------------------- pass 1 over x: per-class column partial sums -
// float4 per thread: one block covers 1024 columns with b128 coalesced loads.
__global__ void partial_sums_kernel(const float* __restrict__ x,
                                    const int* __restrict__ y,
                                    float* __restrict__ part) {
    const int col = (blockIdx.x * 256 + threadIdx.x) * 4;
    const int chunk = blockIdx.y;
    const int r0 = chunk * ROWS_PER_CHUNK;
    const v4f vz = {0.f, 0.f, 0.f, 0.f};
    v4f a0 = vz, a1 = vz, a2 = vz, a3 = vz, a4 = vz;
    for (int r = 0; r < ROWS_PER_CHUNK; ++r) {
        const int row = r0 + r;
        __builtin_prefetch(&x[(size_t)(row + 4) * Dn + col], 0, 3);
        const v4f v = *(const v4f*)(x + (size_t)row * Dn + col);
        const int c = y[row];                     // uniform -> scalar load
        a0 += (c == 0) ? v : vz;
        a1 += (c == 1) ? v : vz;
        a2 += (c == 2) ? v : vz;
        a3 += (c == 3) ? v : vz;
        a4 += (c == 4) ? v : vz;
    }
    float* p = part + (size_t)chunk * TYPEN * Dn + col;
    *(v4f*)(p + 0 * Dn) = a0;
    *(v4f*)(p + 1 * Dn) = a1;
    *(v4f*)(p + 2 * Dn) = a2;
    *(v4f*)(p + 3 * Dn) = a3;
    *(v4f*)(p + 4 * Dn) = a4;
}

// -------------------------------------------- centroids from partial sums -
__global__ void center_kernel(const float* __restrict__ part,
                              const int* __restrict__ counts,
                              float* __restrict__ center) {
    const int idx = blockIdx.x * 256 + threadIdx.x;   // idx = c*Dn + col
    if (idx >= TYPEN * Dn) return;
    float s = 0.f;
    for (int ch = 0; ch < NCHUNK; ++ch) s += part[(size_t)ch * TYPEN * Dn + idx];
    const int c = idx / Dn;
    const float cnt = (float)counts[c];
    center[idx] = s / fmaxf(cnt, 1.0f);
}

// ---- pass 2 over x: WMMA  D[16 rows x 16 classes] = X_tile * Center^T ----
// V_WMMA_F32_16X16X4_F32 fragment layout (ISA 7.12.2):
//   A 16x4 : lane l -> row (l&15), contiguous float2 at k + 2*(l>>4)
//   B 4x16 : lane l -> col (l&15), contiguous float2 of center[n][.] at same k offset
// LDS centroid buffer padded to 6 rows (row 5 = zeros) so every lane has a
// valid unconditional B-fragment address -> no EXEC save/restore in the loop.
__global__ void wmma_pass2_kernel(const float* __restrict__ x,
                                  const float* __restrict__ center,
                                  float* __restrict__ Ct,
                                  float* __restrict__ normx) {
    __shared__ float s_center[(TYPEN + 1) * Dn];      // 48 KB, row 5 zeroed
    const int tid = threadIdx.x;
    for (int i = tid; i < (TYPEN + 1) * Dn; i += 256) {
        float v = 0.f;
        if (i < TYPEN * Dn) v = center[i];
        s_center[i] = v;
    }
    __syncthreads();

    const int wave = tid >> 5;                        // 8 waves / block
    const int lane = tid & 31;
    const int tile = blockIdx.x * 8 + wave;           // 16-row tile id
    const int row0 = tile * 16;
    const int m = lane & 15;                          // row within tile / class col
    const int khalf = (lane >> 4) << 1;               // 0 or 2

    const float* xrow = x + (size_t)(row0 + m) * Dn;
    const float* brow = s_center + (m < TYPEN ? m : TYPEN) * Dn;  // padded zero row

    v8f acc = {0.f, 0.f, 0.f, 0.f, 0.f, 0.f, 0.f, 0.f};
    float sq = 0.f;
    for (int kb = 0; kb < Dn; kb += 32) {
        __builtin_prefetch(xrow + kb + 1024, 0, 3);   // one line-prefetch per 128B/row
#pragma unroll
        for (int ku = 0; ku < 32; ku += 4) {
            const int k0 = kb + ku + khalf;
            v2f a = *(const v2f*)(xrow + k0);
            v2f b = *(const v2f*)(brow + k0);
            sq = fmaf(a.x, a.x, sq);
            sq = fmaf(a.y, a.y, sq);
            acc = __builtin_amdgcn_wmma_f32_16x16x4_f32(
                /*neg_a=*/false, a, /*neg_b=*/false, b,
                /*c_mod=*/(short)0, acc, /*reuse_a=*/false, /*reuse_b=*/false);
        }
    }
    // row norm^2: combine the two K-halves of each row (lanes l and l^16)
    const float tot = sq + __shfl_xor(sq, 16, 32);
    if (lane < 16) normx[row0 + m] = tot;

    // spill D tile: VGPR r, lanes 0-15 -> M=r, lanes 16-31 -> M=8+r (N = lane&15)
    const int Mbase = (lane >> 4) * 8;
#pragma unroll
    for (int r = 0; r < 8; ++r)
        Ct[(size_t)tile * 256 + (Mbase + r) * 16 + m] = acc[r];
}

// ------------------------------------------------------ scalar epilogue ---
__global__ void finalize_kernel(const float* __restrict__ center,
                                const int* __restrict__ counts,
                                const int* __restrict__ y,
                                const float* __restrict__ Ct,
                                const float* __restrict__ normx,
                                const float* __restrict__ cearr,
                                float* __restrict__ out) {
    __shared__ float red[256];
    const int tid = threadIdx.x;

    auto reduceOne = [&](float v) -> float {
        red[tid] = v;
        __syncthreads();
        for (int s = 128; s > 0; s >>= 1) {
            if (tid < s) red[tid] += red[tid + s];
            __syncthreads();
        }
        const float r = red[0];
        __syncthreads();
        return r;
    };

    // one k-loop: centroid row sums, squared norms, 10 pairwise dots
    float rs[TYPEN] = {0.f, 0.f, 0.f, 0.f, 0.f};
    float nc2[TYPEN] = {0.f, 0.f, 0.f, 0.f, 0.f};
    float dot[10] = {0.f, 0.f, 0.f, 0.f, 0.f, 0.f, 0.f, 0.f, 0.f, 0.f};
    for (int k = tid; k < Dn; k += 256) {
        float cv[TYPEN];
#pragma unroll
        for (int c = 0; c < TYPEN; ++c) cv[c] = center[c * Dn + k];
#pragma unroll
        for (int c = 0; c < TYPEN; ++c) { rs[c] += cv[c]; nc2[c] = fmaf(cv[c], cv[c], nc2[c]); }
        int p = 0;
#pragma unroll
        for (int i = 0; i < TYPEN; ++i)
#pragma unroll
            for (int j = i + 1; j < TYPEN; ++j) { dot[p] = fmaf(cv[i], cv[j], dot[p]); ++p; }
    }
    float RS[TYPEN], NC2[TYPEN], DOT[10];
    for (int c = 0; c < TYPEN; ++c) RS[c] = reduceOne(rs[c]);
    for (int c = 0; c < TYPEN; ++c) NC2[c] = reduceOne(nc2[c]);
    for (int p = 0; p < 10; ++p) DOT[p] = reduceOne(dot[p]);

    float normc[TYPEN];
#pragma unroll
    for (int c = 0; c < TYPEN; ++c) normc[c] = sqrtf(NC2[c]);

    // per-row cosine, per-class segment sums + CE sum
    float s0 = 0.f, s1 = 0.f, s2 = 0.f, s3 = 0.f, s4 = 0.f, ces = 0.f;
    for (int row = tid; row < Bn; row += 256) {
        const int c = y[row];
        const float num = Ct[(size_t)(row >> 4) * 256 + (row & 15) * 16 + c];
        const float nx = sqrtf(normx[row]);
        const float nc = (c == 0) ? normc[0] : (c == 1) ? normc[1] : (c == 2) ? normc[2]
                        : (c == 3) ? normc[3] : normc[4];
        const float cosv = num / fmaxf(nx * nc, CEPS);
        s0 += (c == 0) ? cosv : 0.f;
        s1 += (c == 1) ? cosv : 0.f;
        s2 += (c == 2) ? cosv : 0.f;
        s3 += (c == 3) ? cosv : 0.f;
        s4 += (c == 4) ? cosv : 0.f;
        ces += cearr[row];
    }
    float SEG[TYPEN];
    SEG[0] = reduceOne(s0); SEG[1] = reduceOne(s1); SEG[2] = reduceOne(s2);
    SEG[3] = reduceOne(s3); SEG[4] = reduceOne(s4);
    const float CES = reduceOne(ces);

    if (tid == 0) {
        float realn = 0.f, lpsum = 0.f;
        bool mask[TYPEN];
#pragma unroll
        for (int c = 0; c < TYPEN; ++c) {
            mask[c] = (RS[c] != 0.0f);
            realn += mask[c] ? 1.f : 0.f;
            const float cnt = (float)counts[c];
            lpsum += mask[c] ? (SEG[c] / fmaxf(cnt, 1.f)) : 0.f;
        }
        float lp = lpsum / fmaxf(realn, 1.f);
        lp = lp * lp;

        float simsum = 0.f;
        int p = 0;
#pragma unroll
        for (int i = 0; i < TYPEN; ++i)
#pragma unroll
            for (int j = i + 1; j < TYPEN; ++j) {
                simsum += (mask[i] && mask[j])
                              ? DOT[p] / fmaxf(normc[i] * normc[j], CEPS)
                              : 0.f;
                ++p;
            }
        const float npairs = realn * (realn - 1.f) * 0.5f;
        float ln = simsum / fmaxf(npairs, 1.f);
        ln = ln * ln;
        ln = (realn > 1.f) ? ln : 0.f;

        const float cos_loss = -logf((lp + 0.001f) / (ln + 1.001f)) * 0.5f;
        const float ce = CES / (float)Bn;
        out[0] = ce + ((cos_loss < 3.f) ? cos_loss : 0.f);
    }
}

// ------------------------------------------------------------------ host --
extern "C" void kernel_launch(void* const* d_in, const int* in_sizes, int n_in,
                              void* d_out, int out_size, void* d_ws, size_t ws_size,
                              hipStream_t stream) {
    const float* oh = (const float*)d_in[0];   // output_h [Bn, 5]
    const float* th = (const float*)d_in[1];   // target_h [Bn, 5]
    const float* x  = (const float*)d_in[2];   // x        [Bn, Dn]
    float* out = (float*)d_out;

    float* wsf = (float*)d_ws;
    int*   wsi = (int*)d_ws;

    int*   w_y      = wsi + WS_Y;
    float* w_ce     = wsf + WS_CE;
    float* w_part   = wsf + WS_PART;
    int*   w_cnt    = wsi + WS_CNT;
    float* w_center = wsf + WS_CENTER;
    float* w_ct     = wsf + WS_CT;
    float* w_normx  = wsf + WS_NORMX;

    init_kernel<<<1, 32, 0, stream>>>(w_cnt);
    prep_kernel<<<Bn / 256, 256, 0, stream>>>(oh, th, w_y, w_ce, w_cnt);
    partial_sums_kernel<<<dim3(Dn / 1024, NCHUNK), 256, 0, stream>>>(x, w_y, w_part);
    center_kernel<<<(TYPEN * Dn + 255) / 256, 256, 0, stream>>>(w_part, w_cnt, w_center);
    wmma_pass2_kernel<<<Bn / 128, 256, 0, stream>>>(x, w_center, w_ct, w_normx);
    finalize_kernel<<<1, 256, 0, stream>>>(w_center, w_cnt, w_y, w_ct, w_normx, w_ce, out);
}